// InContextRepresentation_30691836297230
// MI455X (gfx1250) — compile-verified
//
#include <hip/hip_runtime.h>

// ---------------------------------------------------------------------------
// MI455X (gfx1250) implementation.
// Memory-bound workload (~65 GFLOP fp32 vs ~270MB minimum HBM traffic) ->
// keep fp32 end-to-end, V_WMMA_F32_16X16X4_F32 for every GEMM, and stage
// tiles into LDS with double-buffered GLOBAL_LOAD_ASYNC_TO_LDS_B128 (ASYNCcnt
// path, no VGPR round-trip) so DMA overlaps WMMA. Tile layouts are template
// parameters so LDS fragment reads lower to ds_load_2addr/b64 with immediate
// offsets. All epilogues (bias / relu / residual / mask / transposed store)
// are fused so each intermediate makes one pass through L2/HBM.
// ---------------------------------------------------------------------------

typedef __attribute__((ext_vector_type(2))) float v2f;
typedef __attribute__((ext_vector_type(8))) float v8f;

struct GStr {
  long aB, aM, aK;   // A[m,k]   = A   + b*aB + m*aM + k*aK
  long bB, bK, bN;   // B[k,n]   = Bm  + b*bB + k*bK + n*bN
  long cB, cM, cN;   // C[m,n]   = C   + b*cB + m*cM + n*cN
  long rB, rM, rN;   // residual (optional)
  long mB;           // mask[b*mB + m] (optional)
};

enum { F_BIAS = 1, F_RELU = 2, F_MASK = 4, F_RES = 8 };

__device__ __forceinline__ v8f wmma_f32_4(v2f a, v2f b, v8f c) {
  // D = A(16x4,f32) * B(4x16,f32) + C(16x16,f32)
  return __builtin_amdgcn_wmma_f32_16x16x4_f32(
      /*neg_a=*/false, a, /*neg_b=*/false, b,
      /*c_mod=*/(short)0, c, /*reuse_a=*/false, /*reuse_b=*/false);
}

// Async 16B copy global -> LDS (tracked by ASYNCcnt, no VGPR round-trip).
// LDS aperture maps addr[31:0] to the LDS offset, so the low 32 bits of the
// generic shared-memory pointer are the per-lane LDS byte address.
__device__ __forceinline__ void async_copy16(const float* g, float* l) {
  const unsigned loff = (unsigned)(unsigned long long)(void*)l;
  asm volatile("global_load_async_to_lds_b128 %0, %1, off"
               : : "v"(loff), "v"(g) : "memory");
}
__device__ __forceinline__ void wait_async0() {
  asm volatile("s_wait_asynccnt 0x0" ::: "memory");
}

// LDS tile layouts (floats). Row pitches chosen for 16B-aligned async writes
// and conflict-free wave32 reads (gcd(pitch,64-banks) small, kh-split lands
// on disjoint bank sets):
//   AROW=1 (aK==1): As[m][k],  pitch 36, 64x32 tile -> 2304 floats
//   AROW=0 (aM==1): At[k][m],  pitch 72, 32x64 tile -> 2304 floats
//   BROW=1 (bN==1): Bs[k][n],  pitch 40, 32x32 tile -> 1280 floats
//   BROW=0 (bK==1): Bt[n][k],  pitch 36, 32x32 tile -> 1152 floats
//
// Generic double-buffered LDS WMMA GEMM.
// Block = 256 threads = 8 waves in a 4(M) x 2(N) grid; block tile 64x32;
// K stepped by 32. Each wave owns one 16x16 C tile.
// WMMA f32 fragment layout (per ISA):
//   A (16x4): lane = m + 16*(k>=2), vgpr j -> K = j + 2*(lane/16)
//   B (4x16): mirrored K mapping, lane%16 carries N
//   C (16x16): row = r + 8*(lane/16), col = lane%16
template <bool AROW, bool BROW>
__global__ __launch_bounds__(256) void k_gemm_wmma(
    const float* __restrict__ A, const float* __restrict__ Bm,
    float* __restrict__ C, const float* __restrict__ bias,
    const float* __restrict__ resid, const float* __restrict__ mask,
    int M, int N, int K, GStr s, int flags) {
  __shared__ float Ash[2][2304];
  __shared__ float Bsh[2][1280];

  const int t    = threadIdx.x;
  const int lane = t & 31;
  const int wave = t >> 5;
  const int wm = wave & 3, wn = wave >> 2;
  const int m0b = blockIdx.x * 64;
  const int n0b = blockIdx.y * 32;
  const int b   = blockIdx.z;
  const int ml  = lane & 15;   // m within wave tile (A) / n within (B,C)
  const int kh  = lane >> 4;   // K-half selector

  const float* Ab = A  + (long)b * s.aB;
  const float* Bb = Bm + (long)b * s.bB;

  auto stage = [&](int d, int k0) {
    if (AROW) {                    // rows contiguous in k
      const int m = t >> 3, kq = (t & 7) * 4;
      async_copy16(Ab + (long)(m0b + m) * s.aM + (k0 + kq),
                   &Ash[d][m * 36 + kq]);
      async_copy16(Ab + (long)(m0b + m + 32) * s.aM + (k0 + kq),
                   &Ash[d][(m + 32) * 36 + kq]);
    } else {                       // aM == 1: columns contiguous -> k-major
      const int k = t >> 4, m4 = (t & 15) * 4;
      async_copy16(Ab + (long)(k0 + k) * s.aK + (m0b + m4),
                   &Ash[d][k * 72 + m4]);
      async_copy16(Ab + (long)(k0 + k + 16) * s.aK + (m0b + m4),
                   &Ash[d][(k + 16) * 72 + m4]);
    }
    if (BROW) {                    // rows contiguous in n
      const int k = t >> 3, nq = (t & 7) * 4;
      async_copy16(Bb + (long)(k0 + k) * s.bK + (n0b + nq),
                   &Bsh[d][k * 40 + nq]);
    } else {                       // bK == 1: columns contiguous -> n-major
      const int n = t >> 3, kq = (t & 7) * 4;
      async_copy16(Bb + (long)(n0b + n) * s.bN + (k0 + kq),
                   &Bsh[d][n * 36 + kq]);
    }
  };

  // Per-lane fragment bases; per-unrolled-iteration offsets are constants so
  // LDS reads fold to ds_load_2addr/b64 with immediate offsets.
  const int arow = wm * 16 + ml;
  const int bcol = wn * 16 + ml;
  const int aBase = AROW ? (arow * 36 + 2 * kh) : (2 * kh * 72 + arow);
  const int bBase = BROW ? (2 * kh * 40 + bcol) : (bcol * 36 + 2 * kh);

  v8f acc = {0.f, 0.f, 0.f, 0.f, 0.f, 0.f, 0.f, 0.f};

  stage(0, 0);
  int buf = 0;
  for (int k0 = 0; k0 < K; k0 += 32) {
    wait_async0();       // this wave's copies for tile `buf` have landed
    __syncthreads();     // everyone's copies landed; prev reads of buf^1 done
    if (k0 + 32 < K) stage(buf ^ 1, k0 + 32);
    const float* Abuf = Ash[buf];
    const float* Bbuf = Bsh[buf];
#pragma unroll
    for (int kk = 0; kk < 32; kk += 4) {
      v2f av, bv;
      if (AROW) {
        av[0] = Abuf[aBase + kk];
        av[1] = Abuf[aBase + kk + 1];
      } else {
        av[0] = Abuf[aBase + kk * 72];
        av[1] = Abuf[aBase + kk * 72 + 72];
      }
      if (BROW) {
        bv[0] = Bbuf[bBase + kk * 40];
        bv[1] = Bbuf[bBase + kk * 40 + 40];
      } else {
        bv[0] = Bbuf[bBase + kk];
        bv[1] = Bbuf[bBase + kk + 1];
      }
      acc = wmma_f32_4(av, bv, acc);
    }
    buf ^= 1;
  }

  // ---- fused epilogue ----
  const int col = n0b + bcol;
  const float bb = (flags & F_BIAS) ? bias[col] : 0.f;
#pragma unroll
  for (int r = 0; r < 8; ++r) {
    const int row = m0b + wm * 16 + r + 8 * kh;
    float v = acc[r] + bb;
    if (flags & F_RES)  v += resid[(long)b * s.rB + (long)row * s.rM + (long)col * s.rN];
    if (flags & F_RELU) v = fmaxf(v, 0.f);
    if (flags & F_MASK) v *= mask[(long)b * s.mB + row];
    C[(long)b * s.cB + (long)row * s.cM + (long)col * s.cN] = v;
  }
}

// Encoder (cols 0..511): 3 embedding gathers + tiny dense projection.
// One block per (b,l) row, 128 threads = one output feature each.
__global__ __launch_bounds__(128) void k_encode(
    const int* __restrict__ xseq, const int* __restrict__ xss,
    const int* __restrict__ x2, const float* __restrict__ xdense, int DIN,
    const float* __restrict__ E_seq, const float* __restrict__ E_ss,
    const float* __restrict__ E_two, const float* __restrict__ Wd,
    const float* __restrict__ bd, const float* __restrict__ mask,
    float* __restrict__ enc) {
  const int row = blockIdx.x;
  const int o = threadIdx.x;   // 0..127
  const float mk = mask[row];
  float* er = enc + (long)row * 640;
  er[o]       = E_seq[(long)xseq[row] * 128 + o] * mk;
  er[128 + o] = E_ss[(long)xss[row] * 128 + o] * mk;
  er[256 + o] = E_two[(long)x2[row] * 128 + o] * mk;
  float acc = bd[o];
  const float* xd = xdense + (long)row * DIN;
  for (int i = 0; i < DIN; ++i) acc += xd[i] * Wd[i * 128 + o];
  er[384 + o] = acc * mk;
}

// dinv[b,c] = rsqrt(sum_r Ah[b,r,c]), Ah = (adj!=0) | (r==c)
__global__ void k_dinv(const float* __restrict__ adj, float* __restrict__ dinv, int L) {
  const int b = blockIdx.y;
  const int c = blockIdx.x * blockDim.x + threadIdx.x;
  if (c >= L) return;
  const float* A = adj + (long)b * L * L;
  float sum = 0.f;
  for (int r = 0; r < L; ++r) {
    const float v = A[(long)r * L + c];
    sum += (v != 0.f || r == c) ? 1.f : 0.f;
  }
  dinv[b * L + c] = rsqrtf(sum);
}

// Nm[b,r,c] = Ah * dinv[b,r] * dinv[b,c]
__global__ void k_normadj(const float* __restrict__ adj,
                          const float* __restrict__ dinv,
                          float* __restrict__ Nm, int L, long total) {
  const long idx = (long)blockIdx.x * blockDim.x + threadIdx.x;
  if (idx >= total) return;
  const int c = (int)(idx % L);
  const long t = idx / L;
  const int r = (int)(t % L);
  const int b = (int)(t / L);
  const float v = adj[idx];
  const float ah = (v != 0.f || r == c) ? 1.f : 0.f;
  Nm[idx] = ah * dinv[b * L + r] * dinv[b * L + c];
}

extern "C" void kernel_launch(void* const* d_in, const int* in_sizes, int n_in,
                              void* d_out, int out_size, void* d_ws, size_t ws_size,
                              hipStream_t stream) {
  (void)in_sizes; (void)n_in; (void)out_size; (void)ws_size;
  const int B = 16, LP = 128, LR = 1024, D = 128, DG = 640, KH = 7;

  const int*   x_pep          = (const int*)d_in[0];
  const int*   x_ss_pep       = (const int*)d_in[1];
  const int*   x_2_pep        = (const int*)d_in[2];
  const float* x_dense_pep    = (const float*)d_in[3];
  const float* x_pretrain_pep = (const float*)d_in[4];
  const int*   x_pro          = (const int*)d_in[5];
  const int*   x_ss_pro       = (const int*)d_in[6];
  const int*   x_2_pro        = (const int*)d_in[7];
  const float* x_dense_pro    = (const float*)d_in[8];
  const float* x_pretrain_pro = (const float*)d_in[9];
  const float* x_edge_pep     = (const float*)d_in[10];
  const float* x_edge_pro     = (const float*)d_in[11];
  const float* mask_pep       = (const float*)d_in[12];
  const float* mask_pro       = (const float*)d_in[13];
  const float* E_seq          = (const float*)d_in[14];
  const float* E_ss           = (const float*)d_in[15];
  const float* E_two          = (const float*)d_in[16];
  const float* W_dpep         = (const float*)d_in[17];
  const float* b_dpep         = (const float*)d_in[18];
  const float* W_dpro         = (const float*)d_in[19];
  const float* b_dpro         = (const float*)d_in[20];
  const float* W_ppep         = (const float*)d_in[21];
  const float* b_ppep         = (const float*)d_in[22];
  const float* W_ppro         = (const float*)d_in[23];
  const float* b_ppro         = (const float*)d_in[24];
  const float* Wg_pep1        = (const float*)d_in[25];
  const float* bg_pep1        = (const float*)d_in[26];
  const float* Wg_pep2        = (const float*)d_in[27];
  const float* bg_pep2        = (const float*)d_in[28];
  const float* Wg_pro1        = (const float*)d_in[29];
  const float* bg_pro1        = (const float*)d_in[30];
  const float* Wg_pro2        = (const float*)d_in[31];
  const float* bg_pro2        = (const float*)d_in[32];
  const float* Wt_pep         = (const float*)d_in[33];
  const float* bt_pep         = (const float*)d_in[34];
  const float* Wt_pro         = (const float*)d_in[35];
  const float* bt_pro         = (const float*)d_in[36];

  float* out = (float*)d_out;

  // Workspace carve-up (floats).
  float* w = (float*)d_ws;
  float* enc_pep = w;  w += (size_t)B * LP * DG;
  float* enc_pro = w;  w += (size_t)B * LR * DG;
  float* Npep    = w;  w += (size_t)B * LP * LP;
  float* Npro    = w;  w += (size_t)B * LR * LR;
  float* dinvPep = w;  w += (size_t)B * LP;
  float* dinvPro = w;  w += (size_t)B * LR;
  float* Tpep    = w;  w += (size_t)B * LP * DG;
  float* Tpro    = w;  w += (size_t)B * LR * DG;
  float* Hpep    = w;  w += (size_t)B * LP * DG;
  float* Hpro    = w;  w += (size_t)B * LR * DG;

  auto gemm = [&](auto kern, const float* A, const float* Bm, float* C,
                  const float* bias, const float* resid, const float* mask,
                  int M, int N, int K, int batch, GStr s, int flags) {
    dim3 grid(M / 64, N / 32, batch);
    kern<<<grid, 256, 0, stream>>>(A, Bm, C, bias, resid, mask, M, N, K, s, flags);
  };

  // ---- Encoders -----------------------------------------------------------
  k_encode<<<B * LP, 128, 0, stream>>>(x_pep, x_ss_pep, x_2_pep, x_dense_pep, 3,
                                       E_seq, E_ss, E_two, W_dpep, b_dpep,
                                       mask_pep, enc_pep);
  k_encode<<<B * LR, 128, 0, stream>>>(x_pro, x_ss_pro, x_2_pro, x_dense_pro, 23,
                                       E_seq, E_ss, E_two, W_dpro, b_dpro,
                                       mask_pro, enc_pro);

  // Pretrain projection: enc[:, 512:640] = (x_pretrain @ W_p + b_p) * mask
  {
    GStr s = {0, 1024, 1,   0, 128, 1,   0, DG, 1,   0, 0, 0,   0};
    gemm(k_gemm_wmma<true, true>, x_pretrain_pep, W_ppep, enc_pep + 512, b_ppep,
         nullptr, mask_pep, B * LP, 128, 1024, 1, s, F_BIAS | F_MASK);
    gemm(k_gemm_wmma<true, true>, x_pretrain_pro, W_ppro, enc_pro + 512, b_ppro,
         nullptr, mask_pro, B * LR, 128, 1024, 1, s, F_BIAS | F_MASK);
  }

  // ---- Normalized adjacency ----------------------------------------------
  k_dinv<<<dim3((LP + 255) / 256, B), 256, 0, stream>>>(x_edge_pep, dinvPep, LP);
  k_dinv<<<dim3((LR + 255) / 256, B), 256, 0, stream>>>(x_edge_pro, dinvPro, LR);
  k_normadj<<<(unsigned)(((long)B * LP * LP) / 256), 256, 0, stream>>>(
      x_edge_pep, dinvPep, Npep, LP, (long)B * LP * LP);
  k_normadj<<<(unsigned)(((long)B * LR * LR) / 256), 256, 0, stream>>>(
      x_edge_pro, dinvPro, Npro, LR, (long)B * LR * LR);

  // ---- GCN blocks ---------------------------------------------------------
  // shared-weight GEMM strides: X[B*L,640] @ W[640,640]
  GStr sw = {0, DG, 1,   0, DG, 1,   0, DG, 1,   0, 0, 0,   0};
  // batched aggregation: out[b,c,d] = sum_r N[b,r,c] * t[b,r,d]  (A = N^T)
  auto aggStr = [&](int L) {
    GStr s = {(long)L * L, 1, (long)L,          // A: N^T layout (aM==1)
              (long)L * DG, DG, 1,              // B: t[b]
              (long)L * DG, DG, 1,              // C
              (long)L * DG, DG, 1,              // residual = enc[b]
              (long)L};                         // mask[b*L + row]
    return s;
  };

  // pep branch
  gemm(k_gemm_wmma<true, true>, enc_pep, Wg_pep1, Tpep, nullptr, nullptr,
       nullptr, B * LP, DG, DG, 1, sw, 0);
  gemm(k_gemm_wmma<false, true>, Npep, Tpep, Hpep, bg_pep1, nullptr, nullptr,
       LP, DG, LP, B, aggStr(LP), F_BIAS | F_RELU);
  gemm(k_gemm_wmma<true, true>, Hpep, Wg_pep2, Tpep, nullptr, nullptr,
       nullptr, B * LP, DG, DG, 1, sw, 0);
  gemm(k_gemm_wmma<false, true>, Npep, Tpep, Hpep, bg_pep2, enc_pep, mask_pep,
       LP, DG, LP, B, aggStr(LP), F_BIAS | F_RES | F_RELU | F_MASK);

  // pro branch
  gemm(k_gemm_wmma<true, true>, enc_pro, Wg_pro1, Tpro, nullptr, nullptr,
       nullptr, B * LR, DG, DG, 1, sw, 0);
  gemm(k_gemm_wmma<false, true>, Npro, Tpro, Hpro, bg_pro1, nullptr, nullptr,
       LR, DG, LR, B, aggStr(LR), F_BIAS | F_RELU);
  gemm(k_gemm_wmma<true, true>, Hpro, Wg_pro2, Tpro, nullptr, nullptr,
       nullptr, B * LR, DG, DG, 1, sw, 0);
  gemm(k_gemm_wmma<false, true>, Npro, Tpro, Hpro, bg_pro2, enc_pro, mask_pro,
       LR, DG, LR, B, aggStr(LR), F_BIAS | F_RES | F_RELU | F_MASK);

  // ---- FC-trans heads (fused transposed store) ----------------------------
  // out[k,b,o,l] = relu(sum_d f[b,l,d] * Wt[k,o,d] + bt[k,o])
  auto fctStr = [&](int L) {
    GStr s = {(long)L * DG, DG, 1,      // A = f[b], m=l, k=d  (aK==1)
              0, 1, (long)DG,           // B = Wt[k]^T : B[d,o] = Wt[o*DG + d] (bK==1)
              (long)D * L, 1, (long)L,  // C: [b, o, l] with row=l, col=o
              0, 0, 0,   0};
    return s;
  };
  float* out_pep = out;
  float* out_pro = out + (size_t)KH * B * D * LP;
  for (int k = 0; k < KH; ++k) {
    gemm(k_gemm_wmma<true, false>, Hpep, Wt_pep + (size_t)k * D * DG,
         out_pep + (size_t)k * B * D * LP, bt_pep + (size_t)k * D, nullptr,
         nullptr, LP, D, DG, B, fctStr(LP), F_BIAS | F_RELU);
    gemm(k_gemm_wmma<true, false>, Hpro, Wt_pro + (size_t)k * D * DG,
         out_pro + (size_t)k * B * D * LR, bt_pro + (size_t)k * D, nullptr,
         nullptr, LR, D, DG, B, fctStr(LR), F_BIAS | F_RELU);
  }
}